// AdaptMarginSVLS_21569325761192
// MI455X (gfx1250) — compile-verified
//
#include <hip/hip_runtime.h>
#include <hip/hip_bf16.h>

typedef __attribute__((ext_vector_type(2))) float v2f;
typedef __attribute__((ext_vector_type(8))) float v8f;

#define NCLS 16
#define HH   2048
#define WW   2048
#define HWPIX (HH * WW)
#define BS   256      // threads per block (8 wave32)
#define PPT  4        // pixels per thread
#define SEG  (BS * PPT)        // 1024 pixels per block (row segment)
#define TILE_W (SEG + 2)       // 1026 labels per tile row (w-1 .. w+SEG)
#define NPARTS ((WW / SEG) * HH)  // 2 * 2048 = 4096 block partials

// Full-wave32 sum via V_WMMA_F32_16X16X4_F32.
// A(16x4): lane m (0-15) holds (K0,K1)=(v_m,0); lane m+16 holds (K2,K3)=(v_{m+16},0).
// B = all ones  =>  D[m,n] = v_m + v_{m+16} for every n.
// Lanes 0-15 each hold D[0..7,n] in their 8 D regs, lanes 16-31 hold D[8..15,n],
// so (sum of 8 D comps) + shfl_xor(.,16) == sum over all 32 lanes, in every lane.
__device__ __forceinline__ float wave_sum_wmma(float v) {
    v2f a; a.x = v;    a.y = 0.0f;
    v2f b; b.x = 1.0f; b.y = 1.0f;
    v8f c = {};
    v8f d = __builtin_amdgcn_wmma_f32_16x16x4_f32(
        /*neg_a=*/false, a, /*neg_b=*/false, b,
        /*c_mod=*/(short)0, c, /*reuse_a=*/false, /*reuse_b=*/false);
    float s = d[0] + d[1] + d[2] + d[3] + d[4] + d[5] + d[6] + d[7];
    s += __shfl_xor(s, 16, 32);
    return s;
}

__global__ __launch_bounds__(BS)
void svls_stage1(const float* __restrict__ x, const int* __restrict__ labels,
                 float2* __restrict__ partials) {
    __shared__ int tile[3][TILE_W + 6];   // +6 pad vs bank patterns
    __shared__ float2 wpart[BS / 32];

    const int t  = threadIdx.x;
    const int bx = blockIdx.x;            // segment index along W
    const int h  = blockIdx.y;            // row
    const int w0 = bx * SEG;

    // Cooperative label tile load: rows h-1..h+1, cols w0-1 .. w0+SEG (zero pad OOB)
    for (int i = t; i < 3 * TILE_W; i += BS) {
        const int r  = i / TILE_W;
        const int cc = i - r * TILE_W;
        const int gh = h - 1 + r;
        const int gw = w0 - 1 + cc;
        int lab = 0;
        if ((unsigned)gh < (unsigned)HH && (unsigned)gw < (unsigned)WW)
            lab = labels[gh * WW + gw];
        tile[r][cc] = lab;
    }
    __syncthreads();

    const int wbase = w0 + t * PPT;
    const size_t pixbase = (size_t)h * WW + wbase;

    // Load 16 channels x 4 pixels with 128-bit coalesced loads
    float xs[NCLS][PPT];
#pragma unroll
    for (int c = 0; c < NCLS; ++c) {
        const float4 v = *(const float4*)(x + (size_t)c * HWPIX + pixbase);
        xs[c][0] = v.x; xs[c][1] = v.y; xs[c][2] = v.z; xs[c][3] = v.w;
    }

    float ce_acc = 0.0f, mg_acc = 0.0f;
#pragma unroll
    for (int p = 0; p < PPT; ++p) {
        const int col = t * PPT + p;      // tile column of (w-1) for this pixel

        // logsumexp over 16 classes
        float xm = xs[0][p];
#pragma unroll
        for (int c = 1; c < NCLS; ++c) xm = fmaxf(xm, xs[c][p]);
        float se = 0.0f;
#pragma unroll
        for (int c = 0; c < NCLS; ++c) se += __expf(xs[c][p] - xm);
        const float lse = xm + __logf(se);

        const int lab = tile[1][col + 1];
        ce_acc += lse - xs[lab][p];

        // 3x3 class histogram, nibble-packed (max count 9 fits in 4 bits)
        unsigned long long hist = 0ull;
#pragma unroll
        for (int r = 0; r < 3; ++r) {
            hist += 1ull << (4 * tile[r][col + 0]);
            hist += 1ull << (4 * tile[r][col + 1]);
            hist += 1ull << (4 * tile[r][col + 2]);
        }
        float ms = 0.0f;
#pragma unroll
        for (int c = 0; c < NCLS; ++c) {
            const float cnt = (float)((hist >> (4 * c)) & 0xFull) * (1.0f / 9.0f);
            ms += fabsf(cnt - xs[c][p]);
        }
        mg_acc += ms;
    }

    // Wave32 reduction on the matrix pipe, then 8-wave combine in LDS
    const float ce_w = wave_sum_wmma(ce_acc);
    const float mg_w = wave_sum_wmma(mg_acc);
    const int wave = t >> 5, lane = t & 31;
    if (lane == 0) wpart[wave] = make_float2(ce_w, mg_w);
    __syncthreads();
    if (t == 0) {
        float a = 0.0f, b = 0.0f;
#pragma unroll
        for (int i = 0; i < BS / 32; ++i) { a += wpart[i].x; b += wpart[i].y; }
        partials[h * gridDim.x + bx] = make_float2(a, b);
    }
}

__global__ __launch_bounds__(BS)
void svls_stage2(const float2* __restrict__ partials, float* __restrict__ out) {
    __shared__ float sce[BS];
    __shared__ float smg[BS];
    const int t = threadIdx.x;
    float ce = 0.0f, mg = 0.0f;
    for (int i = t; i < NPARTS; i += BS) {
        const float2 p = partials[i];
        ce += p.x; mg += p.y;
    }
    sce[t] = ce; smg[t] = mg;
    __syncthreads();
    for (int s = BS / 2; s > 0; s >>= 1) {
        if (t < s) { sce[t] += sce[t + s]; smg[t] += smg[t + s]; }
        __syncthreads();
    }
    if (t == 0) {
        const float ce_mean = sce[0] / (float)HWPIX;
        const float mg_mean = smg[0] / ((float)HWPIX * (float)NCLS);
        out[0] = ce_mean + mg_mean;   // loss  (ALPHA = 1.0)
        out[1] = ce_mean;             // loss_ce
        out[2] = mg_mean;             // loss_margin
    }
}

extern "C" void kernel_launch(void* const* d_in, const int* in_sizes, int n_in,
                              void* d_out, int out_size, void* d_ws, size_t ws_size,
                              hipStream_t stream) {
    const float* x      = (const float*)d_in[0];
    const int*   labels = (const int*)d_in[1];
    float*       out    = (float*)d_out;
    float2*      parts  = (float2*)d_ws;   // NPARTS * 8 bytes = 32 KB

    dim3 grid(WW / SEG, HH);               // (2, 2048)
    svls_stage1<<<grid, BS, 0, stream>>>(x, labels, parts);
    svls_stage2<<<1, BS, 0, stream>>>(parts, out);
}